// HierarchicalGAT_46677704573244
// MI455X (gfx1250) — compile-verified
//
#include <hip/hip_runtime.h>
#include <hip/hip_bf16.h>
#include <stdint.h>

typedef __attribute__((ext_vector_type(2))) float v2f;
typedef __attribute__((ext_vector_type(8))) float v8f;

#define FHID 128
#define NHEAD 4
#define CH 32
#define NEG_SLOPE 0.2f
#define LN_EPS 1e-5f
#define WPAD 132   // LDS row pitch for transposed W (conflict-free for b64 frag reads)

__device__ __forceinline__ float wave_reduce_sum(float v) {
  v += __shfl_xor(v, 1, 32);
  v += __shfl_xor(v, 2, 32);
  v += __shfl_xor(v, 4, 32);
  v += __shfl_xor(v, 8, 32);
  v += __shfl_xor(v, 16, 32);
  return v;
}
__device__ __forceinline__ float group8_reduce_sum(float v) {
  v += __shfl_xor(v, 1, 32);
  v += __shfl_xor(v, 2, 32);
  v += __shfl_xor(v, 4, 32);
  return v;
}
__device__ __forceinline__ float lrelu(float v) {
  return (v > 0.f) ? v : v * NEG_SLOPE;
}

// ---------------------------------------------------------------------------
// Dual GEMM: xl = X@Wl + bl ; xr = X@Wr + br   (X: [n,128], W: [128,128])
// fp32 WMMA 16x16x4. Block = 256 thr = 8 waves; each wave owns 16 rows,
// computes all 8 column tiles for BOTH weight matrices (X read once).
// W staged TRANSPOSED in LDS (pitch 132) so every B fragment is one
// 8B-aligned ds_load_b64 straight into an even register pair (no repacking).
// A frag (16x4 f32): lanes 0-15 hold M=lane, K={0,1}; lanes 16-31 K={2,3}.
// B frag (4x16 f32): lanes 0-15 hold N=lane, K={0,1}; lanes 16-31 K={2,3}.
// C/D (16x16 f32):   VGPR j -> M=j (+8 for lanes 16-31), N=lane&15.
// ---------------------------------------------------------------------------
__global__ __launch_bounds__(256) void gemm_dual_kernel(
    const float* __restrict__ X,
    const float* __restrict__ Wl, const float* __restrict__ bl,
    const float* __restrict__ Wr, const float* __restrict__ br,
    float* __restrict__ xl, float* __restrict__ xr, int n)
{
  __shared__ float sWlT[FHID * WPAD];
  __shared__ float sWrT[FHID * WPAD];
  {
    const float4* gl = (const float4*)Wl;
    const float4* gr = (const float4*)Wr;
    for (int idx = threadIdx.x; idx < FHID * FHID / 4; idx += 256) {
      const int k  = idx >> 5;          // source row (K)
      const int n0 = (idx & 31) * 4;    // source col (N)
      float4 vl = gl[idx];
      float4 vr = gr[idx];
      sWlT[(n0 + 0) * WPAD + k] = vl.x;
      sWlT[(n0 + 1) * WPAD + k] = vl.y;
      sWlT[(n0 + 2) * WPAD + k] = vl.z;
      sWlT[(n0 + 3) * WPAD + k] = vl.w;
      sWrT[(n0 + 0) * WPAD + k] = vr.x;
      sWrT[(n0 + 1) * WPAD + k] = vr.y;
      sWrT[(n0 + 2) * WPAD + k] = vr.z;
      sWrT[(n0 + 3) * WPAD + k] = vr.w;
    }
  }
  __syncthreads();

  const int wv   = threadIdx.x >> 5;
  const int lane = threadIdx.x & 31;
  const int row0 = blockIdx.x * 128 + wv * 16;
  const int c16  = lane & 15;
  const int koff = (lane >> 4) * 2;

  const int r  = row0 + c16;
  const int rc = (r < n) ? r : (n - 1);
  const v2f* __restrict__ xrow2 = (const v2f*)(X + (size_t)rc * FHID);

  v8f accL[8], accR[8];
  const v8f vzero = {0.f, 0.f, 0.f, 0.f, 0.f, 0.f, 0.f, 0.f};
#pragma unroll
  for (int ct = 0; ct < 8; ++ct) { accL[ct] = vzero; accR[ct] = vzero; }

  for (int kt = 0; kt < 32; ++kt) {
    const int k = kt * 4 + koff;
    const v2f a = xrow2[kt * 2 + (koff >> 1)];
#pragma unroll
    for (int ct = 0; ct < 8; ++ct) {
      const int ncol = ct * 16 + c16;
      const v2f bLf = *(const v2f*)&sWlT[ncol * WPAD + k];
      accL[ct] = __builtin_amdgcn_wmma_f32_16x16x4_f32(
          false, a, false, bLf, (short)0, accL[ct], false, false);
      const v2f bRf = *(const v2f*)&sWrT[ncol * WPAD + k];
      accR[ct] = __builtin_amdgcn_wmma_f32_16x16x4_f32(
          false, a, false, bRf, (short)0, accR[ct], false, false);
    }
  }

  const int s8 = (lane >> 4) * 8;
  if (row0 + 16 <= n) {
    // fast path: whole tile in range, straight-line stores
#pragma unroll
    for (int ct = 0; ct < 8; ++ct) {
      const int col = ct * 16 + c16;
      const float blv = bl[col];
      const float brv = br[col];
#pragma unroll
      for (int j = 0; j < 8; ++j) {
        const int row = row0 + s8 + j;
        xl[(size_t)row * FHID + col] = accL[ct][j] + blv;
        xr[(size_t)row * FHID + col] = accR[ct][j] + brv;
      }
    }
  } else {
#pragma unroll
    for (int ct = 0; ct < 8; ++ct) {
      const int col = ct * 16 + c16;
      const float blv = bl[col];
      const float brv = br[col];
#pragma unroll
      for (int j = 0; j < 8; ++j) {
        const int row = row0 + s8 + j;
        if (row < n) {
          xl[(size_t)row * FHID + col] = accL[ct][j] + blv;
          xr[(size_t)row * FHID + col] = accR[ct][j] + brv;
        }
      }
    }
  }
}

// ---------------------------------------------------------------------------
// CSR build (dst-major), once per launch, reused by all 3 layers.
// ---------------------------------------------------------------------------
__global__ void zero_int_kernel(int* p, int n) {
  int t = blockIdx.x * blockDim.x + threadIdx.x;
  if (t < n) p[t] = 0;
}
__global__ void zero_float_kernel(float* p, int n) {
  int t = blockIdx.x * blockDim.x + threadIdx.x;
  if (t < n) p[t] = 0.f;
}
__global__ void deg_kernel(const int* __restrict__ dst, int* __restrict__ deg, int E) {
  int e = blockIdx.x * blockDim.x + threadIdx.x;
  if (e < E) atomicAdd(&deg[dst[e]], 1);
}
__global__ void scan_kernel(const int* __restrict__ deg, int* __restrict__ row_ptr, int n) {
  __shared__ int buf[1024];
  __shared__ int carry;
  const int tid = threadIdx.x;
  if (tid == 0) { carry = 0; row_ptr[0] = 0; }
  __syncthreads();
  for (int base = 0; base < n; base += 1024) {
    int v = (base + tid < n) ? deg[base + tid] : 0;
    buf[tid] = v;
    __syncthreads();
    for (int off = 1; off < 1024; off <<= 1) {
      int t = (tid >= off) ? buf[tid - off] : 0;
      __syncthreads();
      buf[tid] += t;
      __syncthreads();
    }
    if (base + tid < n) row_ptr[base + tid + 1] = buf[tid] + carry;
    __syncthreads();
    if (tid == 0) carry += buf[1023];
    __syncthreads();
  }
}
__global__ void cursor_kernel(const int* __restrict__ row_ptr, int* __restrict__ cur, int n) {
  int t = blockIdx.x * blockDim.x + threadIdx.x;
  if (t < n) cur[t] = row_ptr[t];
}
__global__ void fill_kernel(const int* __restrict__ src, const int* __restrict__ dst,
                            int* __restrict__ cur, int* __restrict__ csr_src, int E) {
  int e = blockIdx.x * blockDim.x + threadIdx.x;
  if (e < E) {
    int d = dst[e];
    int slot = atomicAdd(&cur[d], 1);
    csr_src[slot] = src[e];
  }
}

// ---------------------------------------------------------------------------
// Fused GATv2 attention layer, dst-centric: one wave per node.
// Lane l owns channels [4l, 4l+3] (float4) -> head hq = l>>3, group g = l&7.
// Each edge gather is ONE coalesced b128 load (full 512B row per wave);
// per-head score reduction needs only 3 xor-shuffles (8-lane groups).
// pass1: scores + running max; pass2a: exp + denominator (scores -> exp);
// pass2b: alpha-weighted gather; epilogue: bias + residual + LayerNorm, b128.
// ---------------------------------------------------------------------------
__global__ __launch_bounds__(256) void attn_kernel(
    const float* __restrict__ xl, const float* __restrict__ xr,
    const float* __restrict__ hin,
    const float* __restrict__ att, const float* __restrict__ bo,
    const float* __restrict__ gma, const float* __restrict__ bta,
    const int* __restrict__ row_ptr, const int* __restrict__ csr_src,
    float* __restrict__ scores, float* __restrict__ hout, int n)
{
  const int wv   = threadIdx.x >> 5;
  const int lane = threadIdx.x & 31;
  const int i = blockIdx.x * 8 + wv;
  if (i >= n) return;

  const int hq = lane >> 3;   // my head
  const int g  = lane & 7;    // lane within head group

  const float4* __restrict__ xl4  = (const float4*)xl;
  const float4* __restrict__ xr4  = (const float4*)xr;
  const float4* __restrict__ hin4 = (const float4*)hin;

  const float4 attv = ((const float4*)att)[lane];
  const float4 xrv  = xr4[(size_t)i * 32 + lane];

  const int e0 = row_ptr[i];
  const int e1 = row_ptr[i + 1];

  // ---- self-loop score (kept in registers) ----
  const float4 xself = xl4[(size_t)i * 32 + lane];
  float ssf_mine, mx_mine;
  {
    float p = lrelu(xself.x + xrv.x) * attv.x
            + lrelu(xself.y + xrv.y) * attv.y
            + lrelu(xself.z + xrv.z) * attv.z
            + lrelu(xself.w + xrv.w) * attv.w;
    p = group8_reduce_sum(p);
    ssf_mine = p;
    mx_mine  = p;
  }

  // ---- pass 1: per-edge scores + per-head running max ----
  for (int e = e0; e < e1; ++e) {
    const int s = csr_src[e];
    const float4 xv = xl4[(size_t)s * 32 + lane];
    float p = lrelu(xv.x + xrv.x) * attv.x
            + lrelu(xv.y + xrv.y) * attv.y
            + lrelu(xv.z + xrv.z) * attv.z
            + lrelu(xv.w + xrv.w) * attv.w;
    p = group8_reduce_sum(p);
    mx_mine = fmaxf(mx_mine, p);
    if (g == 0) scores[(size_t)e * 4 + hq] = p;  // lanes 0,8,16,24: 16B contiguous
  }

  // ---- pass 2a: softmax denominator; scores -> exp(score - max) ----
  // lanes process 8 edges x 4 heads per iteration
  const int hh = lane & 3;
  const float mxh = __shfl(mx_mine, hh * 8, 32);  // uniform max of head hh
  float zacc = 0.f;
  for (int base = e0; base < e1; base += 8) {
    const int idx = base + (lane >> 2);
    float av = 0.f;
    if (idx < e1) {
      float sv = scores[(size_t)idx * 4 + hh];
      av = __expf(sv - mxh);
      scores[(size_t)idx * 4 + hh] = av;
    }
    zacc += av;
  }
  zacc += __shfl_xor(zacc, 4, 32);
  zacc += __shfl_xor(zacc, 8, 32);
  zacc += __shfl_xor(zacc, 16, 32);
  // z of MY head: any lane with (lane&3)==hq holds it, e.g. lane hq itself
  const float aself = __expf(ssf_mine - mx_mine);
  const float z_mine = __shfl(zacc, hq, 32) + aself;
  const float invz_mine = 1.f / (z_mine + 1e-16f);

  // ---- pass 2b: alpha-weighted gather-accumulate ----
  float4 acc;
  {
    const float as = aself * invz_mine;
    acc.x = as * xself.x; acc.y = as * xself.y;
    acc.z = as * xself.z; acc.w = as * xself.w;
  }
  for (int e = e0; e < e1; ++e) {
    const int s = csr_src[e];
    const float alpha = scores[(size_t)e * 4 + hq] * invz_mine;
    const float4 xv = xl4[(size_t)s * 32 + lane];
    acc.x += alpha * xv.x;
    acc.y += alpha * xv.y;
    acc.z += alpha * xv.z;
    acc.w += alpha * xv.w;
  }

  // ---- epilogue: bias + residual + LayerNorm(128), all b128 ----
  const float4 bov = ((const float4*)bo)[lane];
  const float4 hv  = hin4[(size_t)i * 32 + lane];
  float4 o;
  o.x = acc.x + bov.x + hv.x;
  o.y = acc.y + bov.y + hv.y;
  o.z = acc.z + bov.z + hv.z;
  o.w = acc.w + bov.w + hv.w;
  const float mu = wave_reduce_sum(o.x + o.y + o.z + o.w) * (1.f / FHID);
  float4 d;
  d.x = o.x - mu; d.y = o.y - mu; d.z = o.z - mu; d.w = o.w - mu;
  const float var =
      wave_reduce_sum(d.x * d.x + d.y * d.y + d.z * d.z + d.w * d.w) * (1.f / FHID);
  const float rinv = rsqrtf(var + LN_EPS);
  const float4 gv = ((const float4*)gma)[lane];
  const float4 bv = ((const float4*)bta)[lane];
  float4 res;
  res.x = gv.x * d.x * rinv + bv.x;
  res.y = gv.y * d.y * rinv + bv.y;
  res.z = gv.z * d.z * rinv + bv.z;
  res.w = gv.w * d.w * rinv + bv.w;
  ((float4*)hout)[(size_t)i * 32 + lane] = res;
}

// ---------------------------------------------------------------------------
// Global mean pool + final projection
// ---------------------------------------------------------------------------
__global__ void pool_kernel(const float* __restrict__ hf, const int* __restrict__ batch,
                            float* __restrict__ pooled, int* __restrict__ cnt, int n)
{
  int t = blockIdx.x * blockDim.x + threadIdx.x;
  int i = t >> 7;
  int j = t & 127;
  if (i < n) {
    int b = batch[i];
    atomicAdd(&pooled[b * FHID + j], hf[(size_t)i * FHID + j]);
    if (j == 0) atomicAdd(&cnt[b], 1);
  }
}

__global__ void outgemm_kernel(const float* __restrict__ pooled, const int* __restrict__ cnt,
                               const float* __restrict__ Wout, const float* __restrict__ bout,
                               float* __restrict__ out)
{
  __shared__ float avg[FHID];
  const int b = blockIdx.x;
  const int j = threadIdx.x;
  float c = (float)cnt[b];
  if (c < 1.f) c = 1.f;
  avg[j] = pooled[b * FHID + j] / c;
  __syncthreads();
  float s = bout[j];
  for (int k = 0; k < FHID; ++k) s += avg[k] * Wout[k * FHID + j];
  out[b * FHID + j] = s;
}

// ---------------------------------------------------------------------------
extern "C" void kernel_launch(void* const* d_in, const int* in_sizes, int n_in,
                              void* d_out, int out_size, void* d_ws, size_t ws_size,
                              hipStream_t stream)
{
  const float* x    = (const float*)d_in[0];
  const int* edge   = (const int*)d_in[1];
  const int* batch  = (const int*)d_in[2];
  const int N = in_sizes[0] / FHID;
  const int E = in_sizes[1] / 2;
  const int Bsz = out_size / FHID;
  const int* src = edge;        // edge_index row 0
  const int* dst = edge + E;    // edge_index row 1
  const float* Wout = (const float*)d_in[27];
  const float* bout = (const float*)d_in[28];

  // workspace carve (256B aligned)
  char* p = (char*)d_ws;
  auto carve = [&](size_t bytes) -> char* {
    char* r = p;
    p += (bytes + 255) & ~(size_t)255;
    return r;
  };
  float* hA     = (float*)carve((size_t)N * FHID * 4);
  float* hB     = (float*)carve((size_t)N * FHID * 4);
  float* xl     = (float*)carve((size_t)N * FHID * 4);
  float* xr     = (float*)carve((size_t)N * FHID * 4);
  float* scores = (float*)carve((size_t)E * 4 * 4);
  int* row_ptr  = (int*)carve((size_t)(N + 1) * 4);
  int* tmp      = (int*)carve((size_t)N * 4);   // deg, then cursor
  int* csr      = (int*)carve((size_t)E * 4);
  float* pooled = (float*)carve((size_t)Bsz * FHID * 4);
  int* cnt      = (int*)carve((size_t)Bsz * 4);
  (void)ws_size; (void)n_in;

  // Build dst-CSR once; topology is shared by all 3 layers.
  zero_int_kernel<<<(N + 255) / 256, 256, 0, stream>>>(tmp, N);
  deg_kernel<<<(E + 255) / 256, 256, 0, stream>>>(dst, tmp, E);
  scan_kernel<<<1, 1024, 0, stream>>>(tmp, row_ptr, N);
  cursor_kernel<<<(N + 255) / 256, 256, 0, stream>>>(row_ptr, tmp, N);
  fill_kernel<<<(E + 255) / 256, 256, 0, stream>>>(src, dst, tmp, csr, E);

  const float* cur = x;
  float* bufs[2] = {hA, hB};
  for (int l = 0; l < 3; ++l) {
    const int base = 3 + 8 * l;
    const float* Wl_ = (const float*)d_in[base + 0];
    const float* bl_ = (const float*)d_in[base + 1];
    const float* Wr_ = (const float*)d_in[base + 2];
    const float* br_ = (const float*)d_in[base + 3];
    const float* at_ = (const float*)d_in[base + 4];
    const float* bo_ = (const float*)d_in[base + 5];
    const float* g_  = (const float*)d_in[base + 6];
    const float* be_ = (const float*)d_in[base + 7];
    float* nxt = bufs[l & 1];
    gemm_dual_kernel<<<(N + 127) / 128, 256, 0, stream>>>(cur, Wl_, bl_, Wr_, br_, xl, xr, N);
    attn_kernel<<<(N + 7) / 8, 256, 0, stream>>>(xl, xr, cur, at_, bo_, g_, be_,
                                                 row_ptr, csr, scores, nxt, N);
    cur = nxt;
  }

  zero_float_kernel<<<(Bsz * FHID + 255) / 256, 256, 0, stream>>>(pooled, Bsz * FHID);
  zero_int_kernel<<<1, 256, 0, stream>>>(cnt, Bsz);
  pool_kernel<<<(N * FHID + 255) / 256, 256, 0, stream>>>(cur, batch, pooled, cnt, N);
  outgemm_kernel<<<Bsz, FHID, 0, stream>>>(pooled, cnt, Wout, bout, (float*)d_out);
}